// CollectNeighbourAverageAndMax_36094905155953
// MI455X (gfx1250) — compile-verified
//
#include <hip/hip_runtime.h>
#include <stdint.h>

// CollectNeighbourAverageAndMax for MI455X (gfx1250).
// V rows, K=32 neighbours, F=64 features. One wave32 per row.
// Gather path: CDNA5 async gather-to-LDS (global_load_async_to_lds_b128, GVS
// form with scalar base), quad-buffered, retired in-order via s_wait_asynccnt.

#define KNN_K 32
#define KNN_F 64
#define WAVES_PER_BLOCK 8
// per-wave LDS: 4 stages * 8 rows * 64 floats = 2048 floats = 8KB
#define WAVE_LDS_FLOATS 2048
#define STAGE_FLOATS 512   // 8 rows * 64 floats

__global__ __launch_bounds__(WAVES_PER_BLOCK * 32)
void CollectNeighbourAverageAndMax_36094905155953_kernel(
    const float* __restrict__ x,
    const int*   __restrict__ idxs,
    float*       __restrict__ out,
    int V)
{
    __shared__ float smem[WAVES_PER_BLOCK * WAVE_LDS_FLOATS]; // 64 KB

    const int lane = (int)(threadIdx.x & 31u);
    const int wib  = (int)(threadIdx.x >> 5);
    // v is wave-uniform; make that visible to the compiler so the guard is a
    // scalar branch and the index loads stay on the SMEM path.
    const int v = __builtin_amdgcn_readfirstlane((int)blockIdx.x * WAVES_PER_BLOCK + wib);
    if (v >= V) return;

    // Whole neighbour-index row is wave-uniform -> scalar loads (s_load_b512 x2).
    const int* __restrict__ irow = idxs + (size_t)v * KNN_K;
    int sidx[KNN_K];
    #pragma unroll
    for (int k = 0; k < KNN_K; ++k) sidx[k] = irow[k];

    const uint64_t xbase = (uint64_t)(uintptr_t)x;
    // One async b128 covers 2 neighbour rows: 16 lanes x 16B per 256B row.
    const uint32_t qo      = (uint32_t)((lane & 15) << 4);  // byte offset in row
    const bool     hiHalf  = (lane >= 16);                  // which row of the pair
    const uint32_t laneLds = (hiHalf ? 256u : 0u) | qo;     // offset in 512B pair block

    // LDS byte address of this wave's staging region (generic ptr low 32 bits
    // == LDS offset on gfx1250).
    const uint32_t waveBase = (uint32_t)(uintptr_t)&smem[wib * WAVE_LDS_FLOATS];

    // Issue ALL 16 async gathers up front (8KB in flight per wave).
    // Op p covers neighbours (2p, 2p+1); no buffer is ever reused, so there is
    // no DS-read vs async-write WAR hazard to drain.
    #pragma unroll
    for (int p = 0; p < 16; ++p) {
        const int      nid  = hiHalf ? sidx[2 * p + 1] : sidx[2 * p]; // v_cndmask
        const uint32_t voff = ((uint32_t)nid << 8) | qo;              // idx*256 + qo
        const uint32_t dst  = waveBase + (uint32_t)(p * 512) + laneLds;
        asm volatile("global_load_async_to_lds_b128 %0, %1, %2"
                     :: "v"(dst), "v"(voff), "s"(xbase)
                     : "memory");
    }

    float2 sum; sum.x = 0.0f; sum.y = 0.0f;
    float2 mx;  mx.x = -__builtin_inff(); mx.y = -__builtin_inff();

    const float2* wavePtr = (const float2*)&smem[wib * WAVE_LDS_FLOATS];

    // Async loads retire in order: stage s (4 ops) is complete once
    // ASYNCcnt <= 4*(3-s).
    #pragma unroll
    for (int s = 0; s < 4; ++s) {
        if      (s == 0) { asm volatile("s_wait_asynccnt 0xc" ::: "memory"); }
        else if (s == 1) { asm volatile("s_wait_asynccnt 0x8" ::: "memory"); }
        else if (s == 2) { asm volatile("s_wait_asynccnt 0x4" ::: "memory"); }
        else             { asm volatile("s_wait_asynccnt 0x0" ::: "memory"); }

        const float2* stagePtr = wavePtr + (size_t)(s * (STAGE_FLOATS / 2));
        #pragma unroll
        for (int r = 0; r < 8; ++r) {
            const float2 val = stagePtr[r * 32 + lane];  // ds_load_2addr_b64 pairs
            sum.x += val.x;
            sum.y += val.y;
            mx.x = fmaxf(mx.x, val.x);
            mx.y = fmaxf(mx.y, val.y);
        }
    }

    // Output row: [ mean(64) | max(64) ], coalesced float2 stores.
    float2* orow = (float2*)(out + (size_t)v * (2 * KNN_F));
    float2 mean; mean.x = sum.x * (1.0f / 32.0f); mean.y = sum.y * (1.0f / 32.0f);
    orow[lane]      = mean;
    orow[32 + lane] = mx;
}

extern "C" void kernel_launch(void* const* d_in, const int* in_sizes, int n_in,
                              void* d_out, int out_size, void* d_ws, size_t ws_size,
                              hipStream_t stream) {
    const float* x    = (const float*)d_in[0];
    const int*   idxs = (const int*)d_in[1];
    float*       out  = (float*)d_out;

    const int V = in_sizes[1] / KNN_K;   // idxs is [V, 32]
    const int blocks = (V + WAVES_PER_BLOCK - 1) / WAVES_PER_BLOCK;

    hipLaunchKernelGGL(CollectNeighbourAverageAndMax_36094905155953_kernel,
                       dim3(blocks), dim3(WAVES_PER_BLOCK * 32), 0, stream,
                       x, idxs, out, V);
}